// SurrogateModel_40673340293394
// MI455X (gfx1250) — compile-verified
//
#include <hip/hip_runtime.h>
#include <hip/hip_bf16.h>
#include <math.h>

typedef __attribute__((ext_vector_type(2))) float v2f;
typedef __attribute__((ext_vector_type(8))) float v8f;

#define N_NODES  100000
#define N_EDGES  1600000
#define IN_DIM   256
#define HID_DIM  512
#define EDGE_HID 64
#define BATCH    4096

// ---------------------------------------------------------------------------
// EdgeConv, algebraically reduced:
//   msg[e][k] = xi*w0[k] + (xj - xi)*w1[k] + b[k]   (xi constant per segment)
//   segment_max over e  ==  evaluate at max(xj) if w1>0 else min(xj)
// so we only need per-node max/min of gathered xj (2 atomics/edge, not 64).
// ---------------------------------------------------------------------------

__global__ void init_minmax_kernel(float* __restrict__ mx, float* __restrict__ mn) {
  int i = blockIdx.x * blockDim.x + threadIdx.x;
  if (i < N_NODES) { mx[i] = -INFINITY; mn[i] = INFINITY; }
}

__global__ void edge_minmax_kernel(const int* __restrict__ ei,
                                   const float* __restrict__ node_x,
                                   float* __restrict__ mx, float* __restrict__ mn) {
  int e = blockIdx.x * blockDim.x + threadIdx.x;
  if (e >= N_EDGES) return;
  int src = ei[e];              // edge_index[0][e]
  int dst = ei[N_EDGES + e];    // edge_index[1][e]
  float xj = node_x[src];
  atomicMax(&mx[dst], xj);      // -> global_atomic_max_num_f32 (IEEE maximumNumber)
  atomicMin(&mn[dst], xj);
}

__global__ void expand_gf_kernel(const float* __restrict__ node_x,
                                 const float* __restrict__ mx,
                                 const float* __restrict__ mn,
                                 const float* __restrict__ Wed,   // [64,2] row-major
                                 const float* __restrict__ bed,   // [64]
                                 float* __restrict__ gf) {        // [N_NODES,64]
  int t = blockIdx.x * blockDim.x + threadIdx.x;
  int node = t >> 6;
  int k    = t & 63;
  if (node >= N_NODES) return;
  float m  = mx[node];
  float w0 = Wed[2 * k];
  float w1 = Wed[2 * k + 1];
  float v  = 0.0f;                       // isolated node: PyG fills 0
  if (m != -INFINITY) {
    float xi = node_x[node];
    float xj = (w1 > 0.0f) ? m : mn[node];
    v = xi * w0 + (xj - xi) * w1 + bed[k];   // same formula as reference msg
  }
  gf[node * 64 + k] = v;
}

// ---------------------------------------------------------------------------
// linear1: h[4096,512] = x[4096,256] @ W1[512,256]^T + b1, fp32 WMMA 16x16x4.
// One wave per 16x16 output tile; 8 waves (256 thr) per block; K loop step 4.
//
// ISA layouts (CDNA5 ISA):
//   A 16x4 f32, 2 VGPRs: lanes 0-15 hold K={0,1}, lanes 16-31 hold K={2,3},
//                        M = lane&15.
//   B 4x16 f32, 2 VGPRs: VGPR v, lanes 0-15 -> K=v, lanes 16-31 -> K=v+2,
//                        N = lane&15  (mirrors C/D row striping).
//   C/D 16x16 f32, 8 VGPRs: VGPR v -> M = v + 8*(lane>=16), N = lane&15.
// ---------------------------------------------------------------------------

__global__ void mlp1_wmma_kernel(const float* __restrict__ X,    // [4096,256]
                                 const float* __restrict__ W1,   // [512,256]
                                 const float* __restrict__ b1,   // [512]
                                 float* __restrict__ H) {        // [4096,512]
  const int wave = threadIdx.x >> 5;
  const int lane = threadIdx.x & 31;
  const int tile = blockIdx.x * 8 + wave;          // 8192 tiles = 256 M x 32 N
  const int tm = tile >> 5;                        // 0..255
  const int tn = tile & 31;                        // 0..31
  const int m0 = tm * 16, n0 = tn * 16;
  const int half = lane >> 4;                      // 0: K lo pair, 1: K hi pair
  const int l16  = lane & 15;

  const float* arow = X  + (size_t)(m0 + l16) * IN_DIM;   // A row (M = l16)
  const float* brow = W1 + (size_t)(n0 + l16) * IN_DIM;   // B col of W1^T = W1 row

  v8f c = {};
  #pragma unroll 4
  for (int k0 = 0; k0 < IN_DIM; k0 += 4) {
    const int k = k0 + 2 * half;
    v2f a = *(const v2f*)(arow + k);   // A[m][k], A[m][k+1]
    v2f b = *(const v2f*)(brow + k);   // B[k][n], B[k+1][n] = W1[n][k..k+1]
    // 8 args: (neg_a, A, neg_b, B, c_mod, C, reuse_a, reuse_b)
    c = __builtin_amdgcn_wmma_f32_16x16x4_f32(false, a, false, b,
                                              (short)0, c, false, false);
  }

  const float bias = b1[n0 + l16];
  #pragma unroll
  for (int v = 0; v < 8; ++v) {
    H[(size_t)(m0 + v + 8 * half) * HID_DIM + (n0 + l16)] = c[v] + bias;
  }
}

// linear2: out[4096] = H @ W2[1,512]^T + b2 — one wave per row, shfl reduce.
__global__ void mlp2_kernel(const float* __restrict__ H,
                            const float* __restrict__ W2,
                            const float* __restrict__ b2,
                            float* __restrict__ out) {
  const int row  = blockIdx.x * 8 + (threadIdx.x >> 5);
  const int lane = threadIdx.x & 31;
  const float* h = H + (size_t)row * HID_DIM;
  float s = 0.0f;
  #pragma unroll 4
  for (int k = lane; k < HID_DIM; k += 32) s += h[k] * W2[k];
  #pragma unroll
  for (int off = 16; off > 0; off >>= 1) s += __shfl_down(s, off, 32);
  if (lane == 0) out[row] = s + b2[0];
}

// ---------------------------------------------------------------------------

extern "C" void kernel_launch(void* const* d_in, const int* in_sizes, int n_in,
                              void* d_out, int out_size, void* d_ws, size_t ws_size,
                              hipStream_t stream) {
  (void)in_sizes; (void)n_in; (void)out_size; (void)ws_size;

  const float* x      = (const float*)d_in[0];  // [4096,256]
  const float* node_x = (const float*)d_in[1];  // [100000,1]
  const int*   eidx   = (const int*)  d_in[2];  // [2,1600000]
  const float* W_edge = (const float*)d_in[3];  // [64,2]
  const float* b_edge = (const float*)d_in[4];  // [64]
  const float* W1     = (const float*)d_in[5];  // [512,256]
  const float* b1     = (const float*)d_in[6];  // [512]
  const float* W2     = (const float*)d_in[7];  // [1,512]
  const float* b2     = (const float*)d_in[8];  // [1]

  float* out_mlp = (float*)d_out;               // [4096]
  float* gf      = out_mlp + BATCH;             // [100000,64]

  float* mx = (float*)d_ws;                     // [N_NODES]
  float* mn = mx + N_NODES;                     // [N_NODES]
  float* H  = mn + N_NODES;                     // [4096,512] = 8 MB

  // EdgeConv path
  init_minmax_kernel<<<(N_NODES + 255) / 256, 256, 0, stream>>>(mx, mn);
  edge_minmax_kernel<<<(N_EDGES + 255) / 256, 256, 0, stream>>>(eidx, node_x, mx, mn);
  expand_gf_kernel<<<(N_NODES * EDGE_HID + 255) / 256, 256, 0, stream>>>(
      node_x, mx, mn, W_edge, b_edge, gf);

  // MLP path (fp32 WMMA GEMM, then dot-product reduction)
  mlp1_wmma_kernel<<<(BATCH / 16) * (HID_DIM / 16) / 8, 256, 0, stream>>>(x, W1, b1, H);
  mlp2_kernel<<<BATCH / 8, 256, 0, stream>>>(H, W2, b2, out_mlp);
}